// Prediction_28552942584104
// MI455X (gfx1250) — compile-verified
//
#include <hip/hip_runtime.h>
#include <hip/hip_bf16.h>

#define HH 512
#define WW 512
#define HWSZ (512 * 512)
#define NB 32
#define KTOP 100
#define CAP 65536
#define THRESH_F 0.01f
#define SCALE_F 4.0f

typedef int v4i __attribute__((ext_vector_type(4)));
typedef __attribute__((address_space(1))) int  g_i32;   // global int
typedef __attribute__((address_space(3))) int  l_i32;   // LDS int
typedef __attribute__((address_space(1))) v4i  g_v4i;   // global int4
typedef __attribute__((address_space(3))) v4i  l_v4i;   // LDS int4

// ---- CDNA5 async global->LDS copy helpers ----
__device__ __forceinline__ void async_load_f32_to_lds(float* lds, const float* g) {
#if defined(__HIP_DEVICE_COMPILE__)
#if __has_builtin(__builtin_amdgcn_global_load_async_to_lds_b32)
  __builtin_amdgcn_global_load_async_to_lds_b32(
      (g_i32*)(void*)g, (l_i32*)(void*)lds, 0, 0);
#else
  asm volatile("global_load_async_to_lds_b32 %0, %1, off"
               :: "v"((unsigned)(unsigned long long)(l_i32*)(void*)lds),
                  "v"((unsigned long long)(const void*)g)
               : "memory");
#endif
#else
  (void)lds; (void)g;
#endif
}

// 16-byte granular staging for the aligned tile interior.
__device__ __forceinline__ void async_load_f32x4_to_lds(float* lds, const float* g) {
#if defined(__HIP_DEVICE_COMPILE__)
#if __has_builtin(__builtin_amdgcn_global_load_async_to_lds_b128)
  __builtin_amdgcn_global_load_async_to_lds_b128(
      (g_v4i*)(void*)g, (l_v4i*)(void*)lds, 0, 0);
#else
  async_load_f32_to_lds(lds + 0, g + 0);
  async_load_f32_to_lds(lds + 1, g + 1);
  async_load_f32_to_lds(lds + 2, g + 2);
  async_load_f32_to_lds(lds + 3, g + 3);
#endif
#else
  (void)lds; (void)g;
#endif
}

__device__ __forceinline__ void wait_async_zero() {
#if defined(__HIP_DEVICE_COMPILE__)
#if __has_builtin(__builtin_amdgcn_s_wait_asynccnt)
  __builtin_amdgcn_s_wait_asynccnt(0);
#else
  asm volatile("s_wait_asynccnt 0" ::: "memory");
#endif
#endif
}

// =====================================================================
// Kernel 1: 3x3 NMS over heatmap, LDS-tiled via async b128/b32 loads;
// emit packed candidates (score_bits<<32 | ~idx) with block aggregation.
// LDS tile: 18 rows x 72 floats (288 B row stride). Interior 64 cols at
// col 4 (16B-aligned in LDS and global); halo columns at col 3 / col 68.
// Clamp-padding == -inf padding for the (pooled == heatmap) test.
// =====================================================================
__global__ __launch_bounds__(256)
void fd_nms_candidates(const float* __restrict__ hm,
                       unsigned long long* __restrict__ cand,
                       int* __restrict__ cnt) {
  constexpr int TW = 64, TH = 16;
  constexpr int LW = 72, LH = TH + 2;  // 72 x 18
  __shared__ float tile[LH * LW];
  __shared__ unsigned long long s_cand[512];
  __shared__ int s_cnt;
  __shared__ int s_base;

  const int tid = threadIdx.x;
  const int b   = blockIdx.z;
  const int tx0 = blockIdx.x * TW;
  const int ty0 = blockIdx.y * TH;
  const float* hmB = hm + (size_t)b * HWSZ;

  if (tid == 0) s_cnt = 0;

  // Interior: 18 rows x 16 float4 chunks, b128 async (all shifts, no divides).
  for (int e = tid; e < LH * 16; e += 256) {
    int r = e >> 4, c = e & 15;
    int gy = ty0 + r - 1; gy = gy < 0 ? 0 : (gy > HH - 1 ? HH - 1 : gy);
    async_load_f32x4_to_lds(&tile[r * LW + 4 + c * 4],
                            hmB + (size_t)gy * WW + tx0 + c * 4);
  }
  // Halo columns: 18 rows x {left,right}, b32 async with x-clamp.
  if (tid < LH * 2) {
    int r = tid >> 1;
    int right = tid & 1;
    int gy = ty0 + r - 1; gy = gy < 0 ? 0 : (gy > HH - 1 ? HH - 1 : gy);
    int gx = right ? (tx0 + TW < WW ? tx0 + TW : WW - 1)
                   : (tx0 > 0 ? tx0 - 1 : 0);
    int lc = right ? 68 : 3;
    async_load_f32_to_lds(&tile[r * LW + lc], hmB + (size_t)gy * WW + gx);
  }
  wait_async_zero();
  __syncthreads();

#pragma unroll
  for (int p = 0; p < 4; ++p) {
    int ly = (tid >> 6) + p * 4;
    int lx = tid & 63;
    int ci = (ly + 1) * LW + (lx + 4);
    float ctr = tile[ci];
    if (ctr > THRESH_F) {
      float nm = fmaxf(
          fmaxf(fmaxf(tile[ci - LW - 1], tile[ci - LW]),
                fmaxf(tile[ci - LW + 1], tile[ci - 1])),
          fmaxf(fmaxf(tile[ci + 1], tile[ci + LW - 1]),
                fmaxf(tile[ci + LW], tile[ci + LW + 1])));
      if (ctr >= nm) {
        unsigned idx = (unsigned)((ty0 + ly) * WW + (tx0 + lx));
        unsigned long long key =
            ((unsigned long long)__float_as_uint(ctr) << 32) |
            (unsigned)(~idx);  // descending sort => ties resolve to min idx
        int pos = atomicAdd(&s_cnt, 1);
        if (pos < 512) s_cand[pos] = key;
      }
    }
  }
  __syncthreads();

  int c2 = s_cnt; if (c2 > 512) c2 = 512;
  if (tid == 0) s_base = atomicAdd(&cnt[b], c2);
  __syncthreads();
  int base = s_base;
  for (int i = tid; i < c2; i += 256) {
    int g = base + i;
    if (g < CAP) cand[(size_t)b * CAP + g] = s_cand[i];
  }
}

// =====================================================================
// Kernel 2: per-batch top-100 via 2-level LDS-histogram radix select,
// bitonic sort of survivors, sparse gather + decode.
// =====================================================================
__global__ __launch_bounds__(1024)
void fd_topk_decode(const float* __restrict__ off,
                    const float* __restrict__ wh,
                    const float* __restrict__ lmk,
                    const unsigned long long* __restrict__ cand,
                    const int* __restrict__ cnt,
                    float* __restrict__ out) {
  __shared__ int s_hist[2048];
  __shared__ unsigned long long s_list[1024];
  __shared__ int s_m, s_binB, s_above;
  __shared__ unsigned s_T;

  const int tid = threadIdx.x;
  const int b = blockIdx.x;
  int N = cnt[b]; if (N > CAP) N = CAP;
  const unsigned long long* cb = cand + (size_t)b * CAP;

  if (N > KTOP) {
    // Pass A: histogram of score bits [30:20]
    for (int i = tid; i < 2048; i += 1024) s_hist[i] = 0;
    __syncthreads();
    for (int i = tid; i < N; i += 1024) {
      unsigned sb = (unsigned)(cb[i] >> 32);
      atomicAdd(&s_hist[(sb >> 20) & 2047], 1);
    }
    __syncthreads();
    if (tid == 0) {
      int cum = 0, bin = 0;
      for (int j = 2047; j >= 0; --j) {
        if (cum + s_hist[j] >= KTOP) { bin = j; break; }
        cum += s_hist[j];
      }
      s_binB = bin; s_above = cum;
    }
    __syncthreads();
    int binB = s_binB, above = s_above;
    // Pass B: refine within bin using bits [19:9]
    for (int i = tid; i < 2048; i += 1024) s_hist[i] = 0;
    __syncthreads();
    for (int i = tid; i < N; i += 1024) {
      unsigned sb = (unsigned)(cb[i] >> 32);
      if (((sb >> 20) & 2047) == (unsigned)binB)
        atomicAdd(&s_hist[(sb >> 9) & 2047], 1);
    }
    __syncthreads();
    if (tid == 0) {
      int cum = above, sbin = 0;
      for (int j = 2047; j >= 0; --j) {
        if (cum + s_hist[j] >= KTOP) { sbin = j; break; }
        cum += s_hist[j];
      }
      s_T = ((unsigned)binB << 20) | ((unsigned)sbin << 9);
    }
  } else {
    if (tid == 0) s_T = 0u;
  }
  if (tid == 0) s_m = 0;
  __syncthreads();

  // Compact survivors (count in [KTOP, KTOP + small sub-bin])
  unsigned T = s_T;
  for (int i = tid; i < N; i += 1024) {
    unsigned long long key = cb[i];
    if ((unsigned)(key >> 32) >= T) {
      int p = atomicAdd(&s_m, 1);
      if (p < 1024) s_list[p] = key;
    }
  }
  __syncthreads();
  int M = s_m; if (M > 1024) M = 1024;
  if (tid >= M) s_list[tid] = 0ull;
  __syncthreads();

  // Bitonic sort 1024 keys, descending (zeros sink to the end)
  for (int k = 2; k <= 1024; k <<= 1) {
    for (int j = k >> 1; j > 0; j >>= 1) {
      int ixj = tid ^ j;
      if (ixj > tid) {
        unsigned long long a = s_list[tid], bb = s_list[ixj];
        bool up = (tid & k) == 0;
        if (up ? (a < bb) : (a > bb)) { s_list[tid] = bb; s_list[ixj] = a; }
      }
      __syncthreads();
    }
  }

  // Decode + sparse gather + write
  if (tid < KTOP) {
    const int o = b * KTOP + tid;
    float* out_ids = out;
    float* out_sc  = out + NB * KTOP;
    float* out_bb  = out + 2 * NB * KTOP;
    float* out_lm  = out + 2 * NB * KTOP + 4 * NB * KTOP;
    if (tid < M) {
      unsigned long long key = s_list[tid];
      float sc = __uint_as_float((unsigned)(key >> 32));
      unsigned idx = ~((unsigned)key);
      int y = (int)(idx >> 9);          // W = 512
      int x = (int)(idx & (WW - 1));
      size_t p2 = (size_t)b * 2 * HWSZ + idx;
      float xs = (float)x + off[p2];
      float ys = (float)y + off[p2 + HWSZ];
      float hw2 = wh[p2] * 0.5f;
      float hh2 = wh[p2 + HWSZ] * 0.5f;
      out_ids[o] = 0.0f;                 // C == 1 -> class id always 0
      out_sc[o]  = sc;
      float* bb = out_bb + (size_t)o * 4;
      bb[0] = (xs - hw2) * SCALE_F;
      bb[1] = (ys - hh2) * SCALE_F;
      bb[2] = (xs + hw2) * SCALE_F;
      bb[3] = (ys + hh2) * SCALE_F;
      size_t pl = (size_t)b * 10 * HWSZ + idx;
      float* lo = out_lm + (size_t)o * 10;
#pragma unroll
      for (int c = 0; c < 5; ++c) {
        lo[2 * c]     = (lmk[pl + (size_t)(2 * c) * HWSZ] + xs) * SCALE_F;
        lo[2 * c + 1] = (lmk[pl + (size_t)(2 * c + 1) * HWSZ] + ys) * SCALE_F;
      }
    } else {
      out_ids[o] = -1.0f;
      out_sc[o]  = -1.0f;
      float* bb = out_bb + (size_t)o * 4;
      bb[0] = bb[1] = bb[2] = bb[3] = -1.0f * SCALE_F;  // reference scales after mask
      float* lo = out_lm + (size_t)o * 10;
#pragma unroll
      for (int c = 0; c < 10; ++c) lo[c] = -1.0f * SCALE_F;
    }
  }
}

extern "C" void kernel_launch(void* const* d_in, const int* in_sizes, int n_in,
                              void* d_out, int out_size, void* d_ws, size_t ws_size,
                              hipStream_t stream) {
  (void)in_sizes; (void)n_in; (void)out_size; (void)ws_size;
  const float* hm  = (const float*)d_in[0];
  const float* off = (const float*)d_in[1];
  const float* wh  = (const float*)d_in[2];
  const float* lmk = (const float*)d_in[3];

  // ws layout: [0,256)   : 32 int candidate counters (zeroed each call)
  //            [256, ...) : 32 * CAP packed u64 candidates
  int* cnt = (int*)d_ws;
  unsigned long long* cand = (unsigned long long*)((char*)d_ws + 256);
  (void)hipMemsetAsync(d_ws, 0, 256, stream);

  dim3 g1(WW / 64, HH / 16, NB);
  fd_nms_candidates<<<g1, dim3(256), 0, stream>>>(hm, cand, cnt);
  fd_topk_decode<<<dim3(NB), dim3(1024), 0, stream>>>(off, wh, lmk, cand, cnt,
                                                      (float*)d_out);
}